// GCNLayer_48060684042940
// MI455X (gfx1250) — compile-verified
//
#include <hip/hip_runtime.h>
#include <hip/hip_bf16.h>

typedef __attribute__((ext_vector_type(2))) float v2f;
typedef __attribute__((ext_vector_type(8))) float v8f;

#define NNODES 100000
#define FEAT   128

// ---------------- degree kernels ----------------
__global__ void k_init_deg(float* __restrict__ deg, int n) {
    int i = blockIdx.x * blockDim.x + threadIdx.x;
    if (i < n) deg[i] = 1.0f;   // self-loop contributes 1
}

__global__ void k_count_deg(const int* __restrict__ col, float* __restrict__ deg, int e) {
    int i = blockIdx.x * blockDim.x + threadIdx.x;
    if (i < e) atomicAdd(&deg[col[i]], 1.0f);
}

// ---------------- GEMM: XW = X (100000x128) * W (128x128), fp32 WMMA ----------------
// One wave computes a 16-row strip across all 128 output columns (8 col tiles).
// W is staged in LDS as k-pair float2s: wpair[(k/2)*128 + n] = {W[k][n], W[k+1][n]}
// -> exactly 64KB, and B-fragment reads are contiguous 8B per lane (conflict-free).
__global__ void __launch_bounds__(256) k_gemm(const float* __restrict__ X,
                                              const float* __restrict__ W,
                                              float* __restrict__ XW,
                                              int n_row_tiles) {
    __shared__ v2f wpair[64 * 128];   // 64 KB
    for (int p = threadIdx.x; p < 64 * 128; p += 256) {
        int k2 = p >> 7;          // k pair index
        int n  = p & 127;
        v2f w;
        w.x = W[(2 * k2) * 128 + n];
        w.y = W[(2 * k2 + 1) * 128 + n];
        wpair[p] = w;
    }
    __syncthreads();

    int wave = threadIdx.x >> 5;
    int lane = threadIdx.x & 31;
    int tile = blockIdx.x * 8 + wave;
    if (tile >= n_row_tiles) return;

    int R     = tile * 16;
    int mrow  = R + (lane & 15);
    int khalf = (lane >> 4) << 1;         // lanes 0-15 -> K=0,1 ; lanes 16-31 -> K=2,3
    int nlo   = lane & 15;

    v8f zero = {};
    v8f c[8];
#pragma unroll
    for (int t = 0; t < 8; ++t) c[t] = zero;

    const float* xrow = X + (size_t)mrow * FEAT;
#pragma unroll 4
    for (int kb = 0; kb < 32; ++kb) {
        int koff = kb * 4 + khalf;
        v2f a = *(const v2f*)(xrow + koff);               // A frag: 16x4 f32
#pragma unroll
        for (int t = 0; t < 8; ++t) {
            v2f bf = wpair[(koff >> 1) * 128 + t * 16 + nlo];  // B frag: 4x16 f32
            c[t] = __builtin_amdgcn_wmma_f32_16x16x4_f32(
                false, a, false, bf, (short)0, c[t], false, false);
        }
    }

    int rstore = R + 8 * (lane >> 4);
#pragma unroll
    for (int t = 0; t < 8; ++t) {
#pragma unroll
        for (int g = 0; g < 8; ++g) {
            XW[(size_t)(rstore + g) * FEAT + t * 16 + nlo] = c[t][g];
        }
    }
}

// ---------------- per-node scale: dis = rsqrt(deg); y = dis * xw; out = 0 ----------------
__global__ void k_scale(const float* __restrict__ deg, float* __restrict__ dis,
                        float* __restrict__ xw, float* __restrict__ out, int total) {
    int i = blockIdx.x * blockDim.x + threadIdx.x;
    if (i >= total) return;
    int node = i >> 7;
    int f    = i & 127;
    float s = __frsqrt_rn(deg[node]);
    xw[i] *= s;            // y
    out[i] = 0.0f;         // zero accumulator (d_out is poisoned by harness)
    if (f == 0) dis[node] = s;
}

// ---------------- per-edge gather + scatter-add: out[c] += y[r] ----------------
// One wave per edge: each lane moves a float4 (128 floats / edge).
__global__ void __launch_bounds__(256) k_scatter(const int* __restrict__ row,
                                                 const int* __restrict__ col,
                                                 const float* __restrict__ y,
                                                 float* __restrict__ out, int e) {
    int gid  = blockIdx.x * blockDim.x + threadIdx.x;
    int edge = gid >> 5;
    int lane = gid & 31;
    if (edge >= e) return;
    int r = row[edge];
    int c = col[edge];
    const float4 v = *(const float4*)(y + (size_t)r * FEAT + lane * 4);
    float* dst = out + (size_t)c * FEAT + lane * 4;
    atomicAdd(dst + 0, v.x);
    atomicAdd(dst + 1, v.y);
    atomicAdd(dst + 2, v.z);
    atomicAdd(dst + 3, v.w);
}

// ---------------- finalize: out = relu(dis_c * (acc + y_c) + b) ----------------
__global__ void k_finalize(const float* __restrict__ dis, const float* __restrict__ y,
                           const float* __restrict__ b, float* __restrict__ out, int total) {
    int i = blockIdx.x * blockDim.x + threadIdx.x;
    if (i >= total) return;
    int node = i >> 7;
    int f    = i & 127;
    float v = dis[node] * (out[i] + y[i]) + b[f];
    out[i] = fmaxf(v, 0.0f);
}

extern "C" void kernel_launch(void* const* d_in, const int* in_sizes, int n_in,
                              void* d_out, int out_size, void* d_ws, size_t ws_size,
                              hipStream_t stream) {
    const float* x  = (const float*)d_in[0];
    const int*   ei = (const int*)d_in[1];    // edge_index [2, E]
    const float* W  = (const float*)d_in[2];
    const float* b  = (const float*)d_in[3];
    float* out = (float*)d_out;

    const int N = NNODES;
    const int E = in_sizes[1] / 2;
    const int total = N * FEAT;                 // 12,800,000

    // workspace layout: xw/y [N*128 f32] | deg [N f32] | dis [N f32]
    float* xw  = (float*)d_ws;
    float* deg = xw + (size_t)N * FEAT;
    float* dis = deg + N;

    const int n_row_tiles = N / 16;             // 6250 (exact)

    k_init_deg <<<(N + 255) / 256, 256, 0, stream>>>(deg, N);
    k_count_deg<<<(E + 255) / 256, 256, 0, stream>>>(ei + E, deg, E);
    k_gemm     <<<(n_row_tiles + 7) / 8, 256, 0, stream>>>(x, W, xw, n_row_tiles);
    k_scale    <<<(total + 255) / 256, 256, 0, stream>>>(deg, dis, xw, out, total);
    k_scatter  <<<(E * 32 + 255) / 256, 256, 0, stream>>>(ei, ei + E, xw, out, E);
    k_finalize <<<(total + 255) / 256, 256, 0, stream>>>(dis, xw, b, out, total);
}